// QuantumGatingPath_74431783239900
// MI455X (gfx1250) — compile-verified
//
#include <hip/hip_runtime.h>

typedef __attribute__((ext_vector_type(2))) float v2f;
typedef __attribute__((ext_vector_type(8))) float v8f;
typedef __attribute__((ext_vector_type(4))) unsigned int u32x4;
typedef __attribute__((ext_vector_type(8))) int i32x8;
typedef __attribute__((ext_vector_type(4))) int i32x4;

#define NQ   10
#define DIM  1024
#define STR  36      // padded LDS row stride (floats); TDM pad: 32 dwords + 4 dwords pad
#define RSTR 33      // staging stride for the reduction (odd -> conflict-free column sums)
#define WPB  4       // waves per block (wave32)

__device__ __forceinline__ unsigned lds_off32(const void* p) {
    // generic pointer to __shared__ is {shared_aperture_hi, lds_byte_offset}
    return (unsigned)(unsigned long long)p;
}

__global__ __launch_bounds__(WPB * 32)
void qsim_kernel(const float* __restrict__ base,
                 const float* __restrict__ params,
                 float* __restrict__ out, int batch)
{
    alignas(16) __shared__ float Amat[32 * STR];         // R0..R4 Kronecker (32x32)
    alignas(16) __shared__ float Bmat[32 * STR];         // R5..R9 Kronecker (32x32)
    alignas(16) __shared__ float Sbuf[WPB][32 * STR];    // per-wave state tile
    alignas(16) __shared__ float Cbuf[WPB][32 * STR];    // per-wave intermediate / staging

    const int tid  = threadIdx.x;
    const int lane = tid & 31;
    const int wave = tid >> 5;
    const int bi   = blockIdx.x * WPB + wave;

    float* S = Sbuf[wave];
    float* C = Cbuf[wave];

    // ---- Stage 1a: one TDM descriptor per wave: DMA the 4KB state into LDS,
    //      hardware-padding 4 dwords after every 32 dwords -> row stride 36 floats.
    if (bi < batch) {
        unsigned long long ga = (unsigned long long)(base + (size_t)bi * DIM);
        unsigned lbase = lds_off32(S);

        u32x4 g0;
        g0[0] = 1u;                                       // count=1 (valid descriptor)
        g0[1] = lbase;                                    // lds_addr [63:32]
        g0[2] = (unsigned)ga;                             // global_addr[31:0]
        g0[3] = (unsigned)(ga >> 32) | 0x80000000u;       // addr[56:32] | type=2 [127:126]

        i32x8 g1;
        g1[0] = (2 << 16)       // data_size = 4 bytes
              | (1 << 20)       // pad_enable
              | (4 << 22)       // pad_interval: 1<<4 * 8B = 128B = 32 dwords
              | (3 << 25);      // pad_amount: 4 dwords (16B) -> stride 36 floats
        g1[1] = (DIM & 0xffff) << 16;                     // tensor_dim0[15:0] @ bits 63:48
        g1[2] = (DIM >> 16);                              // tensor_dim0[31:16]; tensor_dim1=0
        g1[3] = (DIM << 16);                              // tile_dim0=1024 @ bits 127:112
        g1[4] = 0;                                        // tile_dim1=0, tile_dim2=0 (1-D)
        g1[5] = DIM;                                      // tensor_dim0_stride (unused for 1-D)
        g1[6] = 0;
        g1[7] = 0;

        i32x4 gz4 = {0, 0, 0, 0};                         // groups 2/3 unused (<=2D)
        i32x8 gz8 = {0, 0, 0, 0, 0, 0, 0, 0};
        __builtin_amdgcn_tensor_load_to_lds(g0, g1, gz4, gz4, gz8, 0);
    }

    // ---- Stage 0 (once per block, overlapped with the tensor DMA): gate matrices ----
    {
        float c[NQ], s[NQ];
        #pragma unroll
        for (int w = 0; w < NQ; ++w) {
            float h = 0.5f * params[w];
            c[w] = cosf(h);
            s[w] = sinf(h);
        }
        for (int e = tid; e < 1024; e += WPB * 32) {
            int ro = e >> 5;   // output 5-bit index (wire0 = MSB)
            int ri = e & 31;   // input 5-bit index
            float a = 1.0f, b = 1.0f;
            #pragma unroll
            for (int w = 0; w < 5; ++w) {
                int bo = (ro >> (4 - w)) & 1;
                int bi2 = (ri >> (4 - w)) & 1;
                float ra = (bo == bi2) ? c[w]     : (bo ? s[w]     : -s[w]);  // RY=[[c,-s],[s,c]]
                float rb = (bo == bi2) ? c[w + 5] : (bo ? s[w + 5] : -s[w + 5]);
                a *= ra;
                b *= rb;
            }
            Amat[ro * STR + ri] = a;
            Bmat[ro * STR + ri] = b;
        }
    }
    __syncthreads();

    if (bi < batch) {
        __builtin_amdgcn_s_wait_tensorcnt(0);               // state tile resident in LDS

        // WMMA operand lane mapping (16x16x4 f32):
        //  A: lane l -> row (l&15), cols k0..k0+1 with k0 = 2*(l>>4)
        //  B: lane l -> col (l&15), rows k0..k0+1
        //  C/D: vgpr g, lane l -> D[g + 8*(l>>4)][l&15]
        const int ml = lane & 15;
        const int kh = (lane >> 4) << 1;   // 0 or 2
        const int mh = (lane >> 4) << 3;   // 0 or 8

        // ---- Stage 2: C = A * S ----
        #pragma unroll
        for (int Mi = 0; Mi < 32; Mi += 16) {
            #pragma unroll
            for (int Ni = 0; Ni < 32; Ni += 16) {
                v8f acc = {0.f, 0.f, 0.f, 0.f, 0.f, 0.f, 0.f, 0.f};
                #pragma unroll
                for (int kb = 0; kb < 32; kb += 4) {
                    int k0 = kb + kh;
                    v2f a, b;
                    a.x = Amat[(Mi + ml) * STR + k0];
                    a.y = Amat[(Mi + ml) * STR + k0 + 1];
                    b.x = S[(k0    ) * STR + Ni + ml];
                    b.y = S[(k0 + 1) * STR + Ni + ml];
                    acc = __builtin_amdgcn_wmma_f32_16x16x4_f32(
                        false, a, false, b, (short)0, acc, false, false);
                }
                #pragma unroll
                for (int g = 0; g < 8; ++g)
                    C[(Mi + g + mh) * STR + Ni + ml] = acc[g];
            }
        }

        // ---- Stage 3: S = C * B^T   (T[m][n] = sum_k C[m][k] * Bmat[n][k]) ----
        #pragma unroll
        for (int Mi = 0; Mi < 32; Mi += 16) {
            #pragma unroll
            for (int Ni = 0; Ni < 32; Ni += 16) {
                v8f acc = {0.f, 0.f, 0.f, 0.f, 0.f, 0.f, 0.f, 0.f};
                #pragma unroll
                for (int kb = 0; kb < 32; kb += 4) {
                    int k0 = kb + kh;
                    v2f a, b;
                    a.x = C[(Mi + ml) * STR + k0];
                    a.y = C[(Mi + ml) * STR + k0 + 1];
                    b.x = Bmat[(Ni + ml) * STR + k0];      // B^T[k][n] = B[n][k]
                    b.y = Bmat[(Ni + ml) * STR + k0 + 1];
                    acc = __builtin_amdgcn_wmma_f32_16x16x4_f32(
                        false, a, false, b, (short)0, acc, false, false);
                }
                #pragma unroll
                for (int g = 0; g < 8; ++g)
                    S[(Mi + g + mh) * STR + Ni + ml] = acc[g];
            }
        }
        asm volatile("s_wait_dscnt 0x0" ::: "memory");      // final state visible wave-wide

        // ---- Stage 4: CNOT chain == Gray permutation final[b] = S[b ^ (b>>1)] ----
        // Pauli sums, shuffle-free via two gather layouts.
        float sx[NQ], u[NQ];   // sx_i = half X sum; u_i = sum of q over "bit set"
        float T1 = 0.0f;       // trace = ||psi||^2 (circuit is orthogonal)

        {   // layout 1: lane owns b with (b&31)==lane  -> wires 0..4 in-register
            float f1[32], q1[32];
            #pragma unroll
            for (int k = 0; k < 32; ++k) {
                int b = k * 32 + lane;
                int g = b ^ (b >> 1);
                f1[k] = S[(g >> 5) * STR + (g & 31)];
            }
            #pragma unroll
            for (int k = 0; k < 32; ++k) { q1[k] = f1[k] * f1[k]; T1 += q1[k]; }
            #pragma unroll
            for (int i = 0; i < 5; ++i) {
                int km = 1 << (4 - i);
                float s_ = 0.f, uu = 0.f;
                #pragma unroll
                for (int k = 0; k < 32; ++k) {
                    if (!(k & km)) s_ += f1[k] * f1[k | km];   // each pair once
                    else           uu += q1[k];
                }
                sx[i] = s_; u[i] = uu;
            }
        }
        {   // layout 2: lane owns b with (b>>5)==lane -> wires 5..9 in-register
            // gray(b) = gray(lane)*32 ^ (gray5(k) ^ ((lane&1)<<4)) : one row per lane
            float f2[32], q2[32];
            int grl = lane ^ (lane >> 1);                 // 5-bit gray of lane
            int cc  = (lane & 1) << 4;
            const float* rb     = &S[grl * STR];
            const float* bplus  = rb + cc;                // t<16 : t^cc = t+cc
            const float* bminus = rb - cc;                // t>=16: t^cc = t-cc
            #pragma unroll
            for (int k = 0; k < 32; ++k) {
                int t = k ^ (k >> 1);
                f2[k] = (t < 16) ? bplus[t] : bminus[t];
            }
            #pragma unroll
            for (int k = 0; k < 32; ++k) q2[k] = f2[k] * f2[k];
            #pragma unroll
            for (int i = 5; i < NQ; ++i) {
                int km = 1 << (9 - i);
                float s_ = 0.f, uu = 0.f;
                #pragma unroll
                for (int k = 0; k < 32; ++k) {
                    if (!(k & km)) s_ += f2[k] * f2[k | km];
                    else           uu += q2[k];
                }
                sx[i] = s_; u[i] = uu;
            }
        }

        // ---- Stage 5: wave reduction via LDS (21 scalars), coalesced 30-lane store ----
        float* stg = C;                                    // reuse intermediate buffer
        #pragma unroll
        for (int i = 0; i < NQ; ++i) {
            stg[i * RSTR + lane]        = sx[i];           // rows 0..9
            stg[(NQ + i) * RSTR + lane] = u[i];            // rows 10..19
        }
        stg[20 * RSTR + lane] = T1;                        // row 20
        asm volatile("s_wait_dscnt 0x0" ::: "memory");

        float* fin = C + 22 * RSTR;
        if (lane < 21) {
            const float* r0 = &stg[lane * RSTR];
            float red = 0.f;
            #pragma unroll
            for (int j = 0; j < 32; ++j) red += r0[j];
            fin[lane] = red;
        }
        asm volatile("s_wait_dscnt 0x0" ::: "memory");

        if (lane < 30) {
            int i = (lane * 11) >> 5;                      // lane / 3
            int r = lane - 3 * i;
            float T   = fin[20];
            float inv = 1.0f / T;                          // folds skipped normalization
            float v;
            if (r == 0)      v = 2.0f * fin[i] * inv;      // <X_i>
            else if (r == 1) v = 0.0f;                     // <Y_i> == 0 (real state)
            else             v = (T - 2.0f * fin[NQ + i]) * inv;  // <Z_i>
            out[(size_t)bi * (3 * NQ) + lane] = v;
        }
    }
}

extern "C" void kernel_launch(void* const* d_in, const int* in_sizes, int n_in,
                              void* d_out, int out_size, void* d_ws, size_t ws_size,
                              hipStream_t stream) {
    const float* base   = (const float*)d_in[0];   // (32768, 1024) f32
    const float* params = (const float*)d_in[1];   // (10,) f32
    float* out = (float*)d_out;                    // (32768, 30) f32
    (void)d_ws; (void)ws_size; (void)n_in; (void)out_size;

    int batch  = in_sizes[0] / DIM;
    int blocks = (batch + WPB - 1) / WPB;
    qsim_kernel<<<blocks, WPB * 32, 0, stream>>>(base, params, out, batch);
}